// ScaledDotProductAttention_21157008900714
// MI455X (gfx1250) — compile-verified
//
#include <hip/hip_runtime.h>

typedef float v2f __attribute__((ext_vector_type(2)));
typedef float v8f __attribute__((ext_vector_type(8)));

#define BATCH   2
#define HEADS   16
#define SEQ     2048
#define DK      64
#define PSTRIDE (SEQ + 4)          // padded LDS row stride (floats)
#define SCALE   0.125f             // 1/sqrt(64)

// One workgroup = 8 wave32 = 256 threads, handles a 16-query strip for one (b,h).
__global__ __launch_bounds__(256) void sdpa_fwd_kernel(
    const float* __restrict__ Q, const float* __restrict__ Km,
    const float* __restrict__ Vm, float* __restrict__ ctx,
    float* __restrict__ attn)
{
    extern __shared__ float smem[];
    float* sP   = smem;                 // [16][PSTRIDE] score/prob strip
    float* sCtx = smem + 16 * PSTRIDE;  // [4][16][16] PV partial scratch

    const int tq   = blockIdx.x;        // query tile index (16 rows)
    const int q0   = tq * 16;
    const int bh   = blockIdx.y;        // fused batch*head
    const int tid  = threadIdx.x;
    const int lane = tid & 31;
    const int wave = tid >> 5;
    const int lr   = lane & 15;         // lane-in-16
    const int lh   = lane >> 4;         // half-wave select

    const size_t bhoff = (size_t)bh * SEQ * DK;
    const float* Qp = Q  + bhoff + (size_t)q0 * DK;
    const float* Kp = Km + bhoff;
    const float* Vp = Vm + bhoff;

    // ---- preload Q tile as WMMA A-operand (16x4 f32 layout), pre-scaled ----
    // lanes 0-15: M=lr, K = 4*ks + {0,1};  lanes 16-31: M=lr, K = 4*ks + {2,3}
    v2f qa[16];
#pragma unroll
    for (int ks = 0; ks < 16; ++ks) {
        v2f t = *(const v2f*)(Qp + lr * DK + ks * 4 + 2 * lh);
        qa[ks].x = t.x * SCALE;
        qa[ks].y = t.y * SCALE;
    }

    const int ntiles = tq + 1;          // causal: key tiles 0..tq
    const int nkeys  = ntiles * 16;

    // =================== scores = (Q*scale) @ K^T, masked ===================
    for (int t = wave; t < ntiles; t += 8) {
        const int n0 = t * 16;
        if (t + 8 < ntiles)             // prefetch this wave's next key tile
            __builtin_prefetch(Kp + (size_t)(n0 + 128 + lr) * DK, 0, 0);
        v8f acc = {};
#pragma unroll
        for (int ks = 0; ks < 16; ++ks) {
            // B 4x16: lanes 0-15 hold K={4ks,4ks+1}, lanes 16-31 K={4ks+2,4ks+3}
            v2f b = *(const v2f*)(Kp + (size_t)(n0 + lr) * DK + ks * 4 + 2 * lh);
            acc = __builtin_amdgcn_wmma_f32_16x16x4_f32(
                false, qa[ks], false, b, (short)0, acc, false, false);
        }
        const bool diag = (t == tq);
#pragma unroll
        for (int r = 0; r < 8; ++r) {
            const int row = r + lh * 8;     // D layout: M = r (+8 for hi half)
            const int col = n0 + lr;        // N = lane & 15
            float v = acc[r];
            if (diag && lr > row) v = -1.0e30f;   // strict upper triangle
            sP[row * PSTRIDE + col] = v;
        }
    }
    __syncthreads();

    // =================== softmax: wave owns rows {2w, 2w+1} =================
    {
        const int row = 2 * wave + lh;
        float* prow = sP + row * PSTRIDE;
        float m = -1.0e30f;
        for (int c = lr; c < nkeys; c += 16) m = fmaxf(m, prow[c]);
#pragma unroll
        for (int off = 8; off; off >>= 1) m = fmaxf(m, __shfl_xor(m, off, 16));
        float l = 0.0f;
        for (int c = lr; c < nkeys; c += 16) {
            float p = __expf(prow[c] - m);
            prow[c] = p;
            l += p;
        }
#pragma unroll
        for (int off = 8; off; off >>= 1) l += __shfl_xor(l, off, 16);
        const float inv = 1.0f / l;
        for (int c = lr; c < nkeys; c += 16) prow[c] *= inv;
    }
    __syncthreads();

    // ============ write attn strip, coalesced float4, zero tail =============
    {
        float* abase = attn + (size_t)bh * SEQ * SEQ + (size_t)q0 * SEQ;
        const int quads = 16 * SEQ / 4;
        for (int i = tid; i < quads; i += 256) {
            const int row = i >> 9;              // (i*4)/SEQ
            const int col = (i << 2) & (SEQ - 1);
            float4 v;
            if (col < nkeys) v = *(const float4*)(sP + row * PSTRIDE + col);
            else             v = make_float4(0.f, 0.f, 0.f, 0.f);
            *(float4*)(abase + (size_t)row * SEQ + col) = v;
        }
    }

    // =================== context = P @ V (fp32 WMMA) ========================
    {
        const int nt   = wave & 3;          // d-tile (16 cols of 64)
        const int half = wave >> 2;         // K-range half
        const int n0   = nt * 16;
        const int kcnt = nkeys >> 1;        // multiple of 8
        const int kbeg = half * kcnt;
        v8f acc = {};
        for (int k = kbeg; k < kbeg + kcnt; k += 4) {
            const int k0 = k + 2 * lh;
            v2f a;                          // A = P[16 x K] from LDS
            a.x = sP[lr * PSTRIDE + k0];
            a.y = sP[lr * PSTRIDE + k0 + 1];
            v2f b;                          // B = V[K x 16]
            b.x = Vp[(size_t)k0 * DK + n0 + lr];
            b.y = Vp[(size_t)(k0 + 1) * DK + n0 + lr];
            acc = __builtin_amdgcn_wmma_f32_16x16x4_f32(
                false, a, false, b, (short)0, acc, false, false);
        }
        if (half) {
#pragma unroll
            for (int r = 0; r < 8; ++r)
                sCtx[nt * 256 + (r + lh * 8) * 16 + lr] = acc[r];
        }
        __syncthreads();
        if (!half) {
            float* cbase = ctx + bhoff + (size_t)q0 * DK + n0 + lr;
#pragma unroll
            for (int r = 0; r < 8; ++r) {
                const int row = r + lh * 8;
                cbase[(size_t)row * DK] = acc[r] + sCtx[nt * 256 + row * 16 + lr];
            }
        }
    }
}

extern "C" void kernel_launch(void* const* d_in, const int* in_sizes, int n_in,
                              void* d_out, int out_size, void* d_ws, size_t ws_size,
                              hipStream_t stream) {
    (void)in_sizes; (void)n_in; (void)out_size; (void)d_ws; (void)ws_size;
    const float* Q = (const float*)d_in[0];
    const float* K = (const float*)d_in[1];
    const float* V = (const float*)d_in[2];
    // d_in[3] is the boolean causal mask; the kernel applies causality directly.

    float* ctx  = (float*)d_out;
    float* attn = ctx + (size_t)BATCH * HEADS * SEQ * DK;   // outputs concat'd

    const size_t lds_bytes = (16 * (size_t)PSTRIDE + 4 * 256) * sizeof(float);
    dim3 grid(SEQ / 16, BATCH * HEADS);
    hipLaunchKernelGGL(sdpa_fwd_kernel, grid, dim3(256), lds_bytes, stream,
                       Q, K, V, ctx, attn);
}